// LowRankLinear_23424751632974
// MI455X (gfx1250) — compile-verified
//
#include <hip/hip_runtime.h>

// y = x @ (A@B)^T + bias, computed as t = x@B^T (32x64), y = t@A^T (32x8192).
// All f32; uses V_WMMA_F32_16X16X4_F32 on gfx1250 (wave32).
//
// Roofline: total traffic ~6 MB (x 1MB + B 2MB + A 2MB + y 1MB) at 23.3 TB/s
// -> ~0.26 us; FLOPs (67 MF) are negligible. Bandwidth-bound, so f32 WMMA is
// the right precision (zero conversion cost, exact math).

typedef __attribute__((ext_vector_type(2))) float v2f;
typedef __attribute__((ext_vector_type(8))) float v8f;

#define IN_F   8192
#define OUT_F  8192
#define RANK   64
#define BATCH  32

// 64 K-chunks x 8 (2 mt x 4 nt) tiles = 512 waves in phase 1 for enough
// memory-level parallelism to stream 3 MB at full HBM rate.
#define KCHUNKS 64
#define KCHUNK  (IN_F / KCHUNKS)   // 128 -> 32 WMMA k-steps per wave

// ---------------- Phase 0: zero intermediate t [BATCH x RANK] ----------------
__global__ void lora_zero_t(float* __restrict__ t) {
    int i = blockIdx.x * blockDim.x + threadIdx.x;
    if (i < BATCH * RANK) t[i] = 0.0f;
}

// ---------------- Phase 1: t = x @ B^T  (M=32, N=64, K=8192) ----------------
// wave id -> (mt in [0,2), nt in [0,4), kc in [0,KCHUNKS))
// A-operand element (m,k) = x[m][k]; B-operand element (k,n) = B[n][k].
__global__ void __launch_bounds__(256) lora_phase1(const float* __restrict__ x,
                                                   const float* __restrict__ Bw,
                                                   float* __restrict__ t) {
    const int wave = (blockIdx.x * blockDim.x + threadIdx.x) >> 5;
    const int lane = threadIdx.x & 31;

    const int kc = wave % KCHUNKS;
    const int nt = (wave / KCHUNKS) & 3;
    const int mt = wave / (KCHUNKS * 4);

    const int hi    = lane >> 4;    // 0: K=0,1  1: K=2,3  (per 16x4 f32 A layout)
    const int lo    = lane & 15;
    const int kbase = hi * 2;

    // A operand: row m of x; B operand: row n of B (gives B^T(k,n))
    const float* xrow = x  + (size_t)(mt * 16 + lo) * IN_F;
    const float* brow = Bw + (size_t)(nt * 16 + lo) * IN_F;

    v8f c = {};
    const int k0 = kc * KCHUNK;
#pragma unroll 8
    for (int k = k0; k < k0 + KCHUNK; k += 4) {
        v2f a = *(const v2f*)(xrow + k + kbase);   // (m, k+kbase), (m, k+kbase+1)
        v2f b = *(const v2f*)(brow + k + kbase);   // (k+kbase, n), (k+kbase+1, n)
        c = __builtin_amdgcn_wmma_f32_16x16x4_f32(
                /*neg_a=*/false, a, /*neg_b=*/false, b,
                /*c_mod=*/(short)0, c, /*reuse_a=*/false, /*reuse_b=*/false);
    }

    // C/D layout: VGPR i -> M = i + 8*hi, N = lo
#pragma unroll
    for (int i = 0; i < 8; ++i) {
        const int m = mt * 16 + i + 8 * hi;
        const int n = nt * 16 + lo;
        atomicAdd(&t[m * RANK + n], c[i]);         // global_atomic_add_f32
    }
}

// ---------------- Phase 2: y = t @ A^T + bias  (M=32, N=8192, K=64) ----------
// A-operand element (m,k) = t[m][k]; B-operand element (k,n) = A[n][k].
__global__ void __launch_bounds__(256) lora_phase2(const float* __restrict__ t,
                                                   const float* __restrict__ Aw,
                                                   const float* __restrict__ bias,
                                                   float* __restrict__ y) {
    const int wave = (blockIdx.x * blockDim.x + threadIdx.x) >> 5;
    const int lane = threadIdx.x & 31;

    const int nt = wave % (OUT_F / 16);   // 0..511
    const int mt = wave / (OUT_F / 16);   // 0..1

    const int hi    = lane >> 4;
    const int lo    = lane & 15;
    const int kbase = hi * 2;

    const float* trow = t  + (size_t)(mt * 16 + lo) * RANK;
    const float* arow = Aw + (size_t)(nt * 16 + lo) * RANK;

    v8f c = {};
#pragma unroll
    for (int k = 0; k < RANK; k += 4) {
        v2f a = *(const v2f*)(trow + k + kbase);
        v2f b = *(const v2f*)(arow + k + kbase);
        c = __builtin_amdgcn_wmma_f32_16x16x4_f32(
                false, a, false, b, (short)0, c, false, false);
    }

    const float bv = bias[nt * 16 + lo];
#pragma unroll
    for (int i = 0; i < 8; ++i) {
        const int m = mt * 16 + i + 8 * hi;
        const int n = nt * 16 + lo;
        y[(size_t)m * OUT_F + n] = c[i] + bv;
    }
}

// ---------------------------------- launch ----------------------------------
extern "C" void kernel_launch(void* const* d_in, const int* in_sizes, int n_in,
                              void* d_out, int out_size, void* d_ws, size_t ws_size,
                              hipStream_t stream) {
    const float* x    = (const float*)d_in[0];   // [32, 8192]
    const float* A    = (const float*)d_in[1];   // [8192, 64]
    const float* B    = (const float*)d_in[2];   // [64, 8192]  (row r contiguous over k)
    const float* bias = (const float*)d_in[3];   // [8192]
    float*       y    = (float*)d_out;           // [32, 8192]
    float*       t    = (float*)d_ws;            // [32, 64] intermediate

    // t = 0
    lora_zero_t<<<(BATCH * RANK + 255) / 256, 256, 0, stream>>>(t);

    // t += x @ B^T : 2*4*KCHUNKS waves = 512 waves = 16384 threads
    lora_phase1<<<(2 * 4 * KCHUNKS * 32) / 256, 256, 0, stream>>>(x, B, t);

    // y = t @ A^T + bias : 2*512 waves = 1024 waves = 32768 threads
    lora_phase2<<<(2 * (OUT_F / 16) * 32) / 256, 256, 0, stream>>>(t, A, bias, y);
}